// QNN_Amplitude_27290222198802
// MI455X (gfx1250) — compile-verified
//
#include <hip/hip_runtime.h>
#include <cstdint>

typedef float v2f __attribute__((ext_vector_type(2)));
typedef float v8f __attribute__((ext_vector_type(8)));

#define FDIM     4096
#define BROWS    128          // rows per block (8 waves x 16)
#define BK       64           // K chunk
#define LDS_S    68           // padded LDS row stride (floats): conflict-free frags
#define NCHUNK   (FDIM / BK)

// ---------------- setup: build M = sum_j w_j (ReU_j ReU_j^T + ImU_j ImU_j^T) -------------
struct cplx { float re, im; };
__device__ inline cplx cmul(cplx a, cplx b){ return { a.re*b.re - a.im*b.im, a.re*b.im + a.im*b.re }; }

__global__ void qnn_setup_kernel(const float* __restrict__ qw,
                                 const float* __restrict__ W_post,
                                 float* __restrict__ M_out)
{
    if (threadIdx.x != 0 || blockIdx.x != 0) return;
    cplx rot[2][4][2][2];
    for (int l = 0; l < 2; ++l)
        for (int q = 0; q < 4; ++q) {
            float phi = qw[(l*4+q)*3+0], th = qw[(l*4+q)*3+1], om = qw[(l*4+q)*3+2];
            float c = cosf(0.5f*th), s = sinf(0.5f*th);
            float a = 0.5f*(phi+om), b = 0.5f*(phi-om);
            cplx ep = { cosf(a), -sinf(a) };   // e^{-i(phi+om)/2}
            cplx em = { cosf(b),  sinf(b) };   // e^{+i(phi-om)/2}
            rot[l][q][0][0] = {  ep.re*c,  ep.im*c };
            rot[l][q][0][1] = { -em.re*s, -em.im*s };
            rot[l][q][1][0] = {  em.re*s, -em.im*s };   // conj(em)*s
            rot[l][q][1][1] = {  ep.re*c, -ep.im*c };   // conj(ep)*c
        }
    cplx U[16][16];
    for (int i = 0; i < 16; ++i)
        for (int j = 0; j < 16; ++j) U[i][j] = { (i==j) ? 1.f : 0.f, 0.f };
    for (int l = 0; l < 2; ++l) {
        cplx K[16][16];
        for (int a = 0; a < 16; ++a)
            for (int b = 0; b < 16; ++b) {
                cplx p = { 1.f, 0.f };
                for (int q = 0; q < 4; ++q) {
                    int ab = (a >> (3-q)) & 1, bb = (b >> (3-q)) & 1;
                    p = cmul(p, rot[l][q][ab][bb]);
                }
                K[a][b] = p;
            }
        cplx T[16][16];
        for (int i = 0; i < 16; ++i)
            for (int j = 0; j < 16; ++j) {
                cplx acc = { 0.f, 0.f };
                for (int k = 0; k < 16; ++k) {
                    cplx m = cmul(K[i][k], U[k][j]);
                    acc.re += m.re; acc.im += m.im;
                }
                T[i][j] = acc;
            }
        // CNOT-ring permutation: r = (l % 3) + 1; U_new[f[j]] = T[j]
        int r = (l % 3) + 1;
        int f[16]; for (int i = 0; i < 16; ++i) f[i] = i;
        for (int i = 0; i < 4; ++i) {
            int cbit = 1 << (3 - i), tbit = 1 << (3 - ((i + r) & 3));
            for (int j = 0; j < 16; ++j) { int v = f[j]; f[j] = (v & cbit) ? (v ^ tbit) : v; }
        }
        for (int j = 0; j < 16; ++j)
            for (int col = 0; col < 16; ++col) U[f[j]][col] = T[j][col];
    }
    for (int k = 0; k < 16; ++k)
        for (int l2 = 0; l2 < 16; ++l2) {
            float acc = 0.f;
            for (int j = 0; j < 16; ++j)
                acc += W_post[j] * (U[j][k].re*U[j][l2].re + U[j][k].im*U[j][l2].im);
            M_out[k*16 + l2] = acc;
        }
}

// ---------------- main: streaming GEMM (WMMA f32 16x16x4) + quadratic-form epilogue -----
__device__ inline void async_b128(uint32_t lds_off, const void* g) {
    asm volatile("global_load_async_to_lds_b128 %0, %1, off"
                 :: "v"(lds_off), "v"(g) : "memory");
}

__global__ void __launch_bounds__(256)
qnn_main_kernel(const float* __restrict__ x, const float* __restrict__ W_pre,
                const float* __restrict__ b_pre, const float* __restrict__ Mmat,
                const float* __restrict__ b_post, float* __restrict__ out)
{
    __shared__ float xs[2][BROWS * LDS_S];   // x tiles (double buffered)
    __shared__ float wsh[2][16 * LDS_S];     // W_pre tiles (double buffered)
    __shared__ float hs[BROWS * 17];         // h tile for epilogue
    __shared__ float Ms[256];                // quadratic-form matrix

    const int t    = threadIdx.x;
    const int lane = t & 31;
    const int wv   = t >> 5;
    const int r0   = blockIdx.x * BROWS;

    Ms[t] = Mmat[t];

    auto issue = [&](int buf, int k0) {
        #pragma unroll
        for (int i = 0; i < 8; ++i) {                 // 128 rows x 16 b128-segs
            int seg = i * 256 + t;
            int row = seg >> 4, col = seg & 15;
            uint32_t loff = (uint32_t)(uintptr_t)&xs[buf][row * LDS_S + col * 4];
            const float* g = x + (size_t)(r0 + row) * FDIM + k0 + col * 4;
            async_b128(loff, g);
        }
        {                                             // 16 rows x 16 b128-segs of W
            int row = t >> 4, col = t & 15;
            uint32_t loff = (uint32_t)(uintptr_t)&wsh[buf][row * LDS_S + col * 4];
            const float* g = W_pre + (size_t)row * FDIM + k0 + col * 4;
            async_b128(loff, g);
        }
    };

    v8f acc = { 0.f, 0.f, 0.f, 0.f, 0.f, 0.f, 0.f, 0.f };

    issue(0, 0);

    // fragment base offsets (floats): A: lane L -> row L%16, K-off 2*(L/16); B mirrors.
    const int arow = (wv * 16 + (lane & 15)) * LDS_S + ((lane >> 4) << 1);
    const int brow = (lane & 15) * LDS_S + ((lane >> 4) << 1);

    for (int c = 0; c < NCHUNK; ++c) {
        const int buf = c & 1;
        if (c + 1 < NCHUNK) {
            issue(buf ^ 1, (c + 1) * BK);
            asm volatile("s_wait_asynccnt 9" ::: "memory");   // chunk c's 9 loads done
        } else {
            asm volatile("s_wait_asynccnt 0" ::: "memory");
        }
        __syncthreads();
        #pragma unroll
        for (int j = 0; j < BK / 4; ++j) {
            v2f a = *(const v2f*)&xs[buf][arow + j * 4];
            v2f b = *(const v2f*)&wsh[buf][brow + j * 4];
            acc = __builtin_amdgcn_wmma_f32_16x16x4_f32(
                      false, a, false, b, (short)0, acc, false, false);
        }
        __syncthreads();
    }

    // scatter h = D + b_pre to LDS: lane L holds D[M = v + 8*(L/16)][N = L%16]
    const float bp = b_pre[lane & 15];
    const int hf = lane >> 4, nn = lane & 15;
    #pragma unroll
    for (int v = 0; v < 8; ++v) {
        int rl = wv * 16 + hf * 8 + v;
        hs[rl * 17 + nn] = acc[v] + bp;
    }
    __syncthreads();

    // per-row: out = (h^T M h) / (h . h) + b_post
    if (t < BROWS) {
        float hv[16];
        float s = 0.f;
        #pragma unroll
        for (int n = 0; n < 16; ++n) { hv[n] = hs[t * 17 + n]; s += hv[n] * hv[n]; }
        float q = 0.f;
        #pragma unroll
        for (int n = 0; n < 16; ++n) {
            float a = 0.f;
            #pragma unroll
            for (int l = 0; l < 16; ++l) a += Ms[n * 16 + l] * hv[l];
            q += hv[n] * a;
        }
        out[r0 + t] = q / s + b_post[0];
    }
}

extern "C" void kernel_launch(void* const* d_in, const int* in_sizes, int n_in,
                              void* d_out, int out_size, void* d_ws, size_t ws_size,
                              hipStream_t stream) {
    const float* x      = (const float*)d_in[0];
    const float* W_pre  = (const float*)d_in[1];
    const float* b_pre  = (const float*)d_in[2];
    const float* qw     = (const float*)d_in[3];
    const float* W_post = (const float*)d_in[4];
    const float* b_post = (const float*)d_in[5];
    float* Mws = (float*)d_ws;
    float* out = (float*)d_out;

    qnn_setup_kernel<<<1, 32, 0, stream>>>(qw, W_post, Mws);

    const int B = in_sizes[0] / FDIM;  // 32768
    qnn_main_kernel<<<B / BROWS, 256, 0, stream>>>(x, W_pre, b_pre, Mws, b_post, out);
}